// PointPillarScatter_30631706755276
// MI455X (gfx1250) — compile-verified
//
#include <hip/hip_runtime.h>
#include <stdint.h>

// ---------------------------------------------------------------------------
// PointPillarScatter for MI455X (gfx1250).
// Strategy: scatter pillar INDICES into a small L2-resident map (6.9 MB),
// then a single output-stationary gather pass writes the 439 MB of BEV grids
// exactly once with coalesced non-temporal B128 stores. Map tiles are staged
// into LDS with gfx1250 async-to-LDS copies (ASYNCcnt), pillar rows are
// prefetched with global_prefetch.
// ---------------------------------------------------------------------------

namespace pps {
constexpr int B    = 4;
constexpr int NX   = 432;
constexpr int NY   = 496;
constexpr int C    = 64;
constexpr int GRID = NX * NY;          // 214272 cells per batch
constexpr int CELLS = B * GRID;        // 857088 cells total (divisible by 1024)
constexpr int SECTION = CELLS * C;     // 54,853,632 floats per output tensor
constexpr int TPB  = 256;
constexpr int CPT  = 4;                // cells per thread (float4 stores)
constexpr int TILE = TPB * CPT;       // 1024 cells per block
}

typedef float v4f __attribute__((ext_vector_type(4)));
typedef int   v4i __attribute__((ext_vector_type(4)));

// ---- map init: fill index maps with -1 (B128 stores, stays in L2) ----------
__global__ void pps_init_maps(int* __restrict__ maps, int n4) {
  int t = blockIdx.x * blockDim.x + threadIdx.x;
  if (t < n4) {
    v4i m = {-1, -1, -1, -1};
    ((v4i*)maps)[t] = m;
  }
}

// ---- scatter pillar index into map (tiny: one dword per pillar) ------------
__global__ void pps_scatter_idx(const int* __restrict__ coords,
                                int* __restrict__ map, int n) {
  int t = blockIdx.x * blockDim.x + threadIdx.x;
  if (t >= n) return;
  v4i cd = ((const v4i*)coords)[t];                 // (b, z, y, x), z == 0
  int cell = cd.y + cd.z * pps::NX + cd.w;          // z + y*NX + x
  map[cd.x * pps::GRID + cell] = t;
}

// ---- output-stationary gather: one full coalesced NT write pass ------------
__global__ void pps_gather(const float* __restrict__ feats,
                           const int* __restrict__ map,
                           float* __restrict__ out) {
  __shared__ int smem[pps::TILE];
  const int tid = threadIdx.x;
  const int tileBase = blockIdx.x * pps::TILE;      // flat cell index

  // Stage this block's 4 KB map tile into LDS via gfx1250 async copy.
  {
    unsigned lds_addr = (unsigned)(uintptr_t)(&smem[0]) + (unsigned)tid * 16u;
    unsigned goff     = (unsigned)tileBase * 4u + (unsigned)tid * 16u;  // bytes
    asm volatile("global_load_async_to_lds_b128 %0, %1, %2"
                 :: "v"(lds_addr), "v"(goff), "s"(map) : "memory");
  }
#if __has_builtin(__builtin_amdgcn_s_wait_asynccnt)
  __builtin_amdgcn_s_wait_asynccnt(0);
#else
  asm volatile("s_wait_asynccnt 0x0" ::: "memory");
#endif
  __syncthreads();

  v4i idx = ((const v4i*)smem)[tid];                // 4 cells per thread

  const int F0   = tileBase + tid * pps::CPT;
  const int b    = F0 / pps::GRID;
  const int cell = F0 - b * pps::GRID;
  float* obase = out + (size_t)b * (size_t)pps::C * (size_t)pps::GRID + cell;

  // Fast path: all 4 cells empty (the overwhelmingly common case) ->
  // pure streaming zero-fill at full store bandwidth.
  if ((idx.x & idx.y & idx.z & idx.w) < 0) {
    v4f z = {0.f, 0.f, 0.f, 0.f};
#pragma unroll 8
    for (int c = 0; c < pps::C; ++c)
      __builtin_nontemporal_store(z, (v4f*)(obase + (size_t)c * pps::GRID));
    return;
  }

  // Mixed tile: branch-free masked gather from L2-resident feature rows.
  const float m0 = (idx.x >= 0) ? 1.f : 0.f;
  const float m1 = (idx.y >= 0) ? 1.f : 0.f;
  const float m2 = (idx.z >= 0) ? 1.f : 0.f;
  const float m3 = (idx.w >= 0) ? 1.f : 0.f;
  const float* f0 = feats + (size_t)(idx.x < 0 ? 0 : idx.x) * pps::C;
  const float* f1 = feats + (size_t)(idx.y < 0 ? 0 : idx.y) * pps::C;
  const float* f2 = feats + (size_t)(idx.z < 0 ? 0 : idx.z) * pps::C;
  const float* f3 = feats + (size_t)(idx.w < 0 ? 0 : idx.w) * pps::C;

  // Prefetch both 128B halves of each live 256B pillar row (global_prefetch).
  if (idx.x >= 0) { __builtin_prefetch(f0, 0, 1); __builtin_prefetch(f0 + 32, 0, 1); }
  if (idx.y >= 0) { __builtin_prefetch(f1, 0, 1); __builtin_prefetch(f1 + 32, 0, 1); }
  if (idx.z >= 0) { __builtin_prefetch(f2, 0, 1); __builtin_prefetch(f2 + 32, 0, 1); }
  if (idx.w >= 0) { __builtin_prefetch(f3, 0, 1); __builtin_prefetch(f3 + 32, 0, 1); }

#pragma unroll 4
  for (int c = 0; c < pps::C; ++c) {
    v4f v;
    v.x = m0 * f0[c];
    v.y = m1 * f1[c];
    v.z = m2 * f2[c];
    v.w = m3 * f3[c];
    __builtin_nontemporal_store(v, (v4f*)(obase + (size_t)c * pps::GRID));
  }
}

// ---- fallback path (only if workspace is too small) ------------------------
__global__ void pps_zerofill(float* __restrict__ out, int n4) {
  int t = blockIdx.x * blockDim.x + threadIdx.x;
  if (t < n4) {
    v4f z = {0.f, 0.f, 0.f, 0.f};
    __builtin_nontemporal_store(z, ((v4f*)out) + t);
  }
}

__global__ void pps_scatter_direct(const float* __restrict__ feats,
                                   const int* __restrict__ coords,
                                   float* __restrict__ out, int n) {
  int t = blockIdx.x * blockDim.x + threadIdx.x;
  if (t >= n) return;
  v4i cd = ((const v4i*)coords)[t];
  int cell = cd.y + cd.z * pps::NX + cd.w;
  float* obase = out + (size_t)cd.x * (size_t)pps::C * (size_t)pps::GRID + cell;
  const float* f = feats + (size_t)t * pps::C;
#pragma unroll 8
  for (int c = 0; c < pps::C; ++c)
    obase[(size_t)c * pps::GRID] = f[c];
}

// ---------------------------------------------------------------------------
extern "C" void kernel_launch(void* const* d_in, const int* in_sizes, int n_in,
                              void* d_out, int out_size, void* d_ws, size_t ws_size,
                              hipStream_t stream) {
  (void)n_in; (void)out_size;
  const float* tfeat = (const float*)d_in[0];
  const int*   tcrd  = (const int*)d_in[1];
  const float* sfeat = (const float*)d_in[2];
  const int*   scrd  = (const int*)d_in[3];
  float* out = (float*)d_out;

  const int nT = in_sizes[1] / 4;   // 16384 template pillars
  const int nS = in_sizes[3] / 4;   // 65536 search pillars

  const size_t mapBytes = (size_t)2 * pps::CELLS * sizeof(int);

  if (ws_size >= mapBytes) {
    int* mapT = (int*)d_ws;
    int* mapS = mapT + pps::CELLS;

    const int initN4 = (2 * pps::CELLS) / 4;                   // 428544
    pps_init_maps<<<(initN4 + pps::TPB - 1) / pps::TPB, pps::TPB, 0, stream>>>(
        mapT, initN4);

    pps_scatter_idx<<<(nT + pps::TPB - 1) / pps::TPB, pps::TPB, 0, stream>>>(
        tcrd, mapT, nT);
    pps_scatter_idx<<<(nS + pps::TPB - 1) / pps::TPB, pps::TPB, 0, stream>>>(
        scrd, mapS, nS);

    const int nBlocks = pps::CELLS / pps::TILE;                // 837
    pps_gather<<<nBlocks, pps::TPB, 0, stream>>>(tfeat, mapT, out);
    pps_gather<<<nBlocks, pps::TPB, 0, stream>>>(sfeat, mapS, out + pps::SECTION);
  } else {
    // Fallback: zero-fill + direct scatter (no workspace needed).
    const int zeroN4 = (2 * pps::SECTION) / 4;
    pps_zerofill<<<(zeroN4 + pps::TPB - 1) / pps::TPB, pps::TPB, 0, stream>>>(
        out, zeroN4);
    pps_scatter_direct<<<(nT + pps::TPB - 1) / pps::TPB, pps::TPB, 0, stream>>>(
        tfeat, tcrd, out, nT);
    pps_scatter_direct<<<(nS + pps::TPB - 1) / pps::TPB, pps::TPB, 0, stream>>>(
        sfeat, scrd, out + pps::SECTION, nS);
  }
}